// GINConv_ptens_50869592655529
// MI455X (gfx1250) — compile-verified
//
#include <hip/hip_runtime.h>

typedef __attribute__((ext_vector_type(2))) float v2f;
typedef __attribute__((ext_vector_type(8))) float v8f;

#define D_FEAT 64
#define H_FEAT 128

// ---------------------------------------------------------------------------
// Kernel 1: ws[n,d] = (1 + eps) * x[n,d]     (seed for neighbor accumulation)
// ---------------------------------------------------------------------------
__global__ void gin_init(const float* __restrict__ x,
                         const float* __restrict__ eps,
                         float* __restrict__ ws, int total) {
    int i = blockIdx.x * blockDim.x + threadIdx.x;
    if (i < total) {
        ws[i] = (1.0f + eps[0]) * x[i];
    }
}

// ---------------------------------------------------------------------------
// Kernel 2: for each edge (s,d): ws[d] += x[s]; ws[s] += x[d]
// 16 threads per edge, 4 floats per thread, float atomics (HBM/atomic bound).
// ---------------------------------------------------------------------------
__global__ void gin_scatter(const float* __restrict__ x,
                            const long long* __restrict__ ei,
                            float* __restrict__ ws, int n_edges) {
    int idx = blockIdx.x * blockDim.x + threadIdx.x;
    if (idx >= n_edges * 16) return;
    int e = idx >> 4;
    int c = (idx & 15) << 2;                 // feature chunk base (0,4,...,60)
    int s = (int)ei[e];
    int d = (int)ei[n_edges + e];
    const float4 xs = *(const float4*)(x + (size_t)s * D_FEAT + c);
    const float4 xd = *(const float4*)(x + (size_t)d * D_FEAT + c);
    float* wd = ws + (size_t)d * D_FEAT + c;
    float* wsrc = ws + (size_t)s * D_FEAT + c;
    atomicAdd(wd + 0, xs.x); atomicAdd(wd + 1, xs.y);
    atomicAdd(wd + 2, xs.z); atomicAdd(wd + 3, xs.w);
    atomicAdd(wsrc + 0, xd.x); atomicAdd(wsrc + 1, xd.y);
    atomicAdd(wsrc + 2, xd.z); atomicAdd(wsrc + 3, xd.w);
}

// ---------------------------------------------------------------------------
// Kernel 3: y = relu(pre @ W1 + b1) @ W2 + b2  via V_WMMA_F32_16X16X4_F32.
// 4 waves / block, each wave owns 16 node rows.
// LDS: region A (32KB) = W1^T, reused as per-wave h staging after barrier;
//      region B (32KB) = W2^T.  Total = 64KB static.
// f32 16x16x4 fragment layout (wave32):
//   A[16x4]: lane m = lane&15, half = lane>>4; vgpr j holds A[m][2*half+j]
//   B[4x16]: lane n = lane&15;               vgpr j holds B[2*half+j][n]
//   C/D[16x16]: lane n = lane&15; vgpr r holds C[r + 8*half][n]
// ---------------------------------------------------------------------------
__global__ __launch_bounds__(128) void gin_mlp_wmma(
        const float* __restrict__ pre,   // [N,64]  ((1+eps)x + neighbor)
        const float* __restrict__ W1,    // [64,128]
        const float* __restrict__ b1,    // [128]
        const float* __restrict__ W2,    // [128,64]
        const float* __restrict__ b2,    // [64]
        float* __restrict__ y,           // [N,64]
        int n_nodes) {
    __shared__ float sA[H_FEAT * D_FEAT];   // 32KB: W1^T, later h staging
    __shared__ float sW2T[D_FEAT * H_FEAT]; // 32KB: W2^T

    const int tid = threadIdx.x;
    const int wave = tid >> 5;
    const int lane = tid & 31;
    const int m = lane & 15;
    const int half = lane >> 4;

    // Stage W1^T:  sA[n*64 + k] = W1[k*128 + n]
    for (int i = tid; i < D_FEAT * H_FEAT; i += 128) {
        int k = i >> 7, n = i & 127;
        sA[n * D_FEAT + k] = W1[i];
    }
    // Stage W2^T:  sW2T[n*128 + k] = W2[k*64 + n]
    for (int i = tid; i < H_FEAT * D_FEAT; i += 128) {
        int k = i >> 6, n = i & 63;
        sW2T[n * H_FEAT + k] = W2[i];
    }
    __syncthreads();

    const int node0 = (blockIdx.x * 4 + wave) * 16;   // wave-uniform
    const bool active = (node0 < n_nodes);            // wave-uniform branch
    const int row = active ? ((node0 + m < n_nodes) ? (node0 + m) : (n_nodes - 1))
                           : 0;

    // ---- GEMM1: h[16x128] = pre_tile[16x64] @ W1 + b1, then ReLU ----
    v8f acc1[8];
    if (active) {
        #pragma unroll
        for (int t = 0; t < 8; ++t) {
            float bv = b1[t * 16 + m];
            #pragma unroll
            for (int r = 0; r < 8; ++r) acc1[t][r] = bv;
        }
        #pragma unroll
        for (int kk = 0; kk < 16; ++kk) {
            const int kbase = kk * 4 + half * 2;
            v2f a = *(const v2f*)(pre + (size_t)row * D_FEAT + kbase);
            #pragma unroll
            for (int t = 0; t < 8; ++t) {
                v2f b = *(const v2f*)(sA + (t * 16 + m) * D_FEAT + kbase);
                acc1[t] = __builtin_amdgcn_wmma_f32_16x16x4_f32(
                    false, a, false, b, (short)0, acc1[t], false, false);
            }
        }
        #pragma unroll
        for (int t = 0; t < 8; ++t)
            #pragma unroll
            for (int r = 0; r < 8; ++r)
                acc1[t][r] = fmaxf(acc1[t][r], 0.0f);
    }
    __syncthreads();   // everyone done reading W1^T; region A becomes h buffer

    // Stage h tile (C/D layout -> row-major) into this wave's LDS slice.
    float* hbuf = sA + wave * (16 * H_FEAT);
    if (active) {
        #pragma unroll
        for (int t = 0; t < 8; ++t)
            #pragma unroll
            for (int r = 0; r < 8; ++r)
                hbuf[(r + 8 * half) * H_FEAT + t * 16 + m] = acc1[t][r];
    }
    __syncthreads();

    // ---- GEMM2: y_tile[16x64] = h[16x128] @ W2 + b2 ----
    v8f acc2[4];
    if (active) {
        #pragma unroll
        for (int t = 0; t < 4; ++t) {
            float bv = b2[t * 16 + m];
            #pragma unroll
            for (int r = 0; r < 8; ++r) acc2[t][r] = bv;
        }
        #pragma unroll
        for (int kk = 0; kk < 32; ++kk) {
            const int kbase = kk * 4 + half * 2;
            v2f a = *(const v2f*)(hbuf + m * H_FEAT + kbase);
            #pragma unroll
            for (int t = 0; t < 4; ++t) {
                v2f b = *(const v2f*)(sW2T + (t * 16 + m) * H_FEAT + kbase);
                acc2[t] = __builtin_amdgcn_wmma_f32_16x16x4_f32(
                    false, a, false, b, (short)0, acc2[t], false, false);
            }
        }
        // Store: D[mr][nc] lives in acc2[t][r] with mr = r+8*half, nc = t*16+m
        #pragma unroll
        for (int t = 0; t < 4; ++t) {
            #pragma unroll
            for (int r = 0; r < 8; ++r) {
                int nr = node0 + r + 8 * half;
                if (nr < n_nodes) y[(size_t)nr * D_FEAT + t * 16 + m] = acc2[t][r];
            }
        }
    }
}

// ---------------------------------------------------------------------------
extern "C" void kernel_launch(void* const* d_in, const int* in_sizes, int n_in,
                              void* d_out, int out_size, void* d_ws, size_t ws_size,
                              hipStream_t stream) {
    const float*      x   = (const float*)d_in[0];
    const float*      W1  = (const float*)d_in[1];
    const float*      b1  = (const float*)d_in[2];
    const float*      W2  = (const float*)d_in[3];
    const float*      b2  = (const float*)d_in[4];
    const float*      eps = (const float*)d_in[5];
    const long long*  ei  = (const long long*)d_in[6];
    float* out = (float*)d_out;
    float* ws  = (float*)d_ws;   // [N, 64] f32 accumulator (12.8 MB)

    const int N = in_sizes[0] / D_FEAT;     // 50000
    const int E = in_sizes[6] / 2;          // 800000

    // K1: ws = (1+eps)*x
    int total = N * D_FEAT;
    gin_init<<<(total + 255) / 256, 256, 0, stream>>>(x, eps, ws, total);

    // K2: neighbor scatter (float atomics)
    int sthreads = E * 16;
    gin_scatter<<<(sthreads + 255) / 256, 256, 0, stream>>>(x, ei, ws, E);

    // K3: WMMA MLP, 64 nodes per block (4 waves x 16)
    int blocks = (N + 63) / 64;
    gin_mlp_wmma<<<blocks, 128, 0, stream>>>(ws, W1, b1, W2, b2, out, N);
}